// BEVDetectionLoss_60327110639965
// MI455X (gfx1250) — compile-verified
//
#include <hip/hip_runtime.h>
#include <hip/hip_bf16.h>

// ---------------------------------------------------------------------------
// BEV detection loss, restructured:
//   cls_loss = (sum_all softplus(z) - sum_{unique (b,cell,lbl)} z) / (B*CELLS)
//   box_loss = sum_{unique marked cells} smoothL1(pred - gt_winner) / (count+1e-6)
// Only cls_logits (84MB) is streamed -> HBM-bound, ~3.6us floor @ 23.3 TB/s.
// softplus uses raw v_exp_f32 / v_log_f32 (arg to log is provably in [1,2],
// so no subnormal guards / compensated ln2 split needed).
// WMMA f32 16x16x4 with an all-ones B matrix is a 64-wide exact f32
// accumulator on the matrix pipe (gfx1250-specific path).
// ---------------------------------------------------------------------------

#define KXMIN   (-51.2f)
#define KRES    (0.2f)
#define KBEVW   512
#define KCELLS  (512 * 512)
#define KC      10
#define KD      7
#define KB      8
#define KN      128

#define LOG2E_F 1.44269504088896340736f
#define LN2_F   0.693147180559945309417f

typedef __attribute__((ext_vector_type(2))) float v2f;
typedef __attribute__((ext_vector_type(8))) float v8f;

__device__ __forceinline__ float softplus_bce(float z) {
    // max(z,0) + log(1 + exp(-|z|)) : stable BCE-with-target-0.
    // exp(-|z|) = exp2(-log2e*|z|)  -> v_mul + v_exp_f32
    // log(1+e)  = log2(1+e) * ln2   -> v_add + v_log_f32, arg in [1,2]
    float e  = __builtin_amdgcn_exp2f(-LOG2E_F * fabsf(z));
    float l2 = __builtin_amdgcn_logf(1.0f + e);
    return fmaf(l2, LN2_F, fmaxf(z, 0.0f));
}

// ---------------------------------------------------------------------------
// Kernel 1: streaming softplus reduction over cls_logits via WMMA accumulate.
// Writes one partial per block into d_ws. Uniform trip counts keep EXEC all-1
// at every v_wmma (ISA requirement).
// ---------------------------------------------------------------------------
__global__ __launch_bounds__(256) void bev_splus_reduce(
    const float* __restrict__ cls, long long total4,
    float* __restrict__ partials) {

    const long long nthreads = (long long)gridDim.x * blockDim.x;
    const long long tid = (long long)blockIdx.x * blockDim.x + threadIdx.x;
    const long long iters = total4 / nthreads;          // scalar-uniform
    const long long rem   = total4 - iters * nthreads;  // handled w/o wmma

    const float4* __restrict__ p4 = (const float4*)cls;

    v2f bones; bones[0] = 1.0f; bones[1] = 1.0f;  // B = all ones (4x16)
    v8f acc0 = {};                                 // independent accumulators
    v8f acc1 = {};                                 // (no D->SRC2 serialization)

    for (long long i = 0; i < iters; ++i) {
        float4 v = p4[tid + i * nthreads];         // coalesced 16B/lane
        v2f a0, a1;
        a0[0] = softplus_bce(v.x); a0[1] = softplus_bce(v.y);
        a1[0] = softplus_bce(v.z); a1[1] = softplus_bce(v.w);
        // D[m,n] = sum_k A[m,k]*1 + C[m,n]  -> exact f32 accumulation
        acc0 = __builtin_amdgcn_wmma_f32_16x16x4_f32(
                   false, a0, false, bones, (short)0, acc0, false, false);
        acc1 = __builtin_amdgcn_wmma_f32_16x16x4_f32(
                   false, a1, false, bones, (short)0, acc1, false, false);
    }

    // Every column of D is identical; take column 0 only (lanes 0 and 16).
    float s = acc0[0] + acc0[1] + acc0[2] + acc0[3]
            + acc0[4] + acc0[5] + acc0[6] + acc0[7]
            + acc1[0] + acc1[1] + acc1[2] + acc1[3]
            + acc1[4] + acc1[5] + acc1[6] + acc1[7];
    const int lane = threadIdx.x & 31;            // wave32
    if ((lane & 15) != 0) s = 0.0f;

    // Remainder float4s (none for this problem size) - plain adds, divergent OK.
    if (tid < rem) {
        float4 v = p4[iters * nthreads + tid];
        s += softplus_bce(v.x) + softplus_bce(v.y)
           + softplus_bce(v.z) + softplus_bce(v.w);
    }

    __shared__ float red[256];
    red[threadIdx.x] = s;
    __syncthreads();
    for (int off = 128; off > 0; off >>= 1) {
        if ((int)threadIdx.x < off) red[threadIdx.x] += red[threadIdx.x + off];
        __syncthreads();
    }
    if (threadIdx.x == 0) partials[blockIdx.x] = red[0];
}

// ---------------------------------------------------------------------------
// Kernel 2: single block, 1024 threads = one thread per gt entry (8x128).
// Dedupe (b,cell) and (b,cell,lbl) keys in LDS, gather scattered logits /
// box_preds, fold in kernel-1 partials, emit [total, cls_loss, box_loss].
// ---------------------------------------------------------------------------
__global__ __launch_bounds__(1024) void bev_gt_finalize(
    const float* __restrict__ cls, const float* __restrict__ box,
    const float* __restrict__ gtb, const int* __restrict__ gtl,
    const float* __restrict__ gtm,
    const float* __restrict__ partials, int numPartials,
    const float* __restrict__ clsTail, int tailCount,
    float* __restrict__ out) {

    const int t = threadIdx.x;            // t = b*128 + n
    __shared__ int   s_cell[KB * KN];
    __shared__ int   s_cls [KB * KN];
    __shared__ float r0[KB * KN], r1[KB * KN], r2[KB * KN], r3[KB * KN];

    const float x = gtb[t * KD + 0];
    const float y = gtb[t * KD + 1];
    const int   lbl = gtl[t];
    const float msk = gtm[t];
    const bool valid = (msk > 0.5f) && (lbl >= 0) &&
                       (x >= KXMIN) && (x <= -KXMIN) &&
                       (y >= KXMIN) && (y <= -KXMIN);

    int gx = (int)((x - KXMIN) / KRES); gx = min(max(gx, 0), KBEVW - 1);
    int gy = (int)((y - KXMIN) / KRES); gy = min(max(gy, 0), KBEVW - 1);
    const int b = t >> 7;
    const int cellkey = b * KCELLS + gy * KBEVW + gx;         // flat [B,CELLS]
    const int clskey  = cellkey * KC + max(lbl, 0);           // flat [B,CELLS,C]

    s_cell[t] = valid ? cellkey : -(t + 1);   // unique negative sentinels
    s_cls [t] = valid ? clskey  : -(t + 1);
    __syncthreads();

    bool clsRep = valid;   // first occurrence of (b,cell,lbl): subtract z once
    bool boxWin = valid;   // last  occurrence of (b,cell): scatter winner
    if (valid) {
        const int ck = clskey, ce = cellkey;
        for (int u = 0; u < KB * KN; ++u) {
            if (u < t && s_cls [u] == ck) clsRep = false;
            if (u > t && s_cell[u] == ce) boxWin = false;
        }
    }

    float corr = 0.0f, num = 0.0f, cnt = 0.0f;
    if (clsRep) corr = cls[(long long)clskey];
    if (boxWin) {
        cnt = 1.0f;
        const long long base = (long long)cellkey * KD;
        #pragma unroll
        for (int d = 0; d < KD; ++d) {
            const float diff = box[base + d] - gtb[t * KD + d];
            const float ad = fabsf(diff);
            num += (ad < 1.0f) ? (0.5f * diff * diff) : (ad - 0.5f);
        }
    }

    float sp = (t < numPartials) ? partials[t] : 0.0f;
    if (t < tailCount) sp += softplus_bce(clsTail[t]);  // n%4 leftovers

    r0[t] = corr; r1[t] = num; r2[t] = cnt; r3[t] = sp;
    __syncthreads();
    for (int off = 512; off > 0; off >>= 1) {
        if (t < off) {
            r0[t] += r0[t + off]; r1[t] += r1[t + off];
            r2[t] += r2[t + off]; r3[t] += r3[t + off];
        }
        __syncthreads();
    }
    if (t == 0) {
        const float inv = 1.0f / (float)((long long)KB * KCELLS);
        const float cls_loss = (r3[0] - r0[0]) * inv;
        const float box_loss = r1[0] / (r2[0] + 1e-6f);
        out[0] = cls_loss + box_loss;   // total (weights are 1.0)
        out[1] = cls_loss;
        out[2] = box_loss;
    }
}

// ---------------------------------------------------------------------------
extern "C" void kernel_launch(void* const* d_in, const int* in_sizes, int n_in,
                              void* d_out, int out_size, void* d_ws, size_t ws_size,
                              hipStream_t stream) {
    const float* cls = (const float*)d_in[0];   // [8, 262144, 10] f32
    const float* box = (const float*)d_in[1];   // [8, 262144, 7]  f32
    const float* gtb = (const float*)d_in[2];   // [8, 128, 7]     f32
    const int*   gtl = (const int*)d_in[3];     // [8, 128]        int
    const float* gtm = (const float*)d_in[4];   // [8, 128]        f32
    float* out = (float*)d_out;                 // [3]
    float* partials = (float*)d_ws;

    const long long n = (long long)in_sizes[0]; // 20,971,520
    const long long total4 = n >> 2;
    const int tail = (int)(n - (total4 << 2));

    const int blocks = 512, threads = 256;      // 4096 waves, 40 float4/thread
    bev_splus_reduce<<<blocks, threads, 0, stream>>>(cls, total4, partials);
    bev_gt_finalize<<<1, 1024, 0, stream>>>(cls, box, gtb, gtl, gtm,
                                            partials, blocks,
                                            cls + (total4 << 2), tail, out);
}